// GATWrapper_17317308137493
// MI455X (gfx1250) — compile-verified
//
#include <hip/hip_runtime.h>

#define NEG_SLOPE 0.2f

typedef __attribute__((ext_vector_type(2))) float v2f;
typedef __attribute__((ext_vector_type(8))) float v8f;

// ---- order-preserving float<->int encoding for atomic segment-max ----
__device__ __forceinline__ int enc_f(float f) {
    int b = __float_as_int(f);
    return (b >= 0) ? b : (b ^ 0x7FFFFFFF);
}
__device__ __forceinline__ float dec_f(int m) {
    int b = (m >= 0) ? m : (m ^ 0x7FFFFFFF);
    return __int_as_float(b);
}
#define EMAX_INIT ((int)0x807FFFFF)   // enc(-inf)

__device__ __forceinline__ float leaky(float v) {
    return v > 0.0f ? v : NEG_SLOPE * v;
}

// ---------------------------------------------------------------------
// 0) init: out <- bias broadcast, emax <- enc(-inf), denom <- 0
// ---------------------------------------------------------------------
__global__ void gat_init(float* __restrict__ out, const float* __restrict__ bias,
                         int out_n, int* __restrict__ emax, float* __restrict__ denom,
                         int nh) {
    int i = blockIdx.x * blockDim.x + threadIdx.x;
    if (i < out_n) out[i] = bias[i & 127];
    if (i < nh) { emax[i] = EMAX_INIT; denom[i] = 0.0f; }
}

// ---------------------------------------------------------------------
// 1) xp = x @ lin_w^T  via V_WMMA_F32_16X16X4_F32
//    block = 256 thr = 8 waves; block computes rows [16b,16b+16) x all 128 cols
//    wave w computes the 16x16 tile at cols [16w, 16w+16)
// ---------------------------------------------------------------------
__global__ void __launch_bounds__(256)
gat_gemm_wmma(const float* __restrict__ x, const float* __restrict__ w,
              float* __restrict__ xp, int n_nodes) {
    __shared__ float wlds[128 * 128];   // 64 KB: full weight matrix
    __shared__ float xtile[16 * 128];   // 8 KB: A tile

    const int tid  = threadIdx.x;
    const int wave = tid >> 5;
    const int lane = tid & 31;
    const int half = lane >> 4;
    const int l16  = lane & 15;
    const int row0 = blockIdx.x * 16;
    const int col0 = wave * 16;

    // cooperative stage of W and the 16x128 x-tile
    for (int i = tid; i < 128 * 128; i += 256) wlds[i] = w[i];
    for (int i = tid; i < 16 * 128; i += 256) {
        int r = row0 + (i >> 7);
        xtile[i] = (r < n_nodes) ? x[(size_t)r * 128 + (i & 127)] : 0.0f;
    }
    __syncthreads();

    // A fragment rows / B fragment cols for this lane
    const float* arow = &xtile[l16 * 128];         // A row m = row0 + l16
    const float* bcol = &wlds[(col0 + l16) * 128]; // B[k][n] = w[n][k]

    v8f acc = {};
    for (int k0 = 0; k0 < 128; k0 += 4) {
        // A 16x4 layout: lanes 0-15 -> K = k0,k0+1 ; lanes 16-31 -> K = k0+2,k0+3
        v2f a, b;
        int ka = k0 + 2 * half;
        a.x = arow[ka];
        a.y = arow[ka + 1];
        // B 4x16 layout (row-striped, half-wave K split) -> same contiguous pair
        b.x = bcol[ka];
        b.y = bcol[ka + 1];
#if __has_builtin(__builtin_amdgcn_wmma_f32_16x16x4_f32)
        acc = __builtin_amdgcn_wmma_f32_16x16x4_f32(
            /*neg_a=*/false, a, /*neg_b=*/false, b,
            /*c_mod=*/(short)0, acc, /*reuse_a=*/false, /*reuse_b=*/false);
#else
        // scalar fallback (should not be taken on gfx1250)
        acc[0] += a.x * b.x + a.y * b.y;
#endif
    }

    // D 16x16 f32: VGPR j -> (M = j + 8*half, N = lane%16)
    #pragma unroll
    for (int j = 0; j < 8; ++j) {
        int r = row0 + j + 8 * half;
        if (r < n_nodes) xp[(size_t)r * 128 + col0 + l16] = acc[j];
    }
}

// ---------------------------------------------------------------------
// 2) per-node attention logits  a_s[n,h], a_d[n,h]
// ---------------------------------------------------------------------
__global__ void gat_logits(const float* __restrict__ xp,
                           const float* __restrict__ att_src,
                           const float* __restrict__ att_dst,
                           float* __restrict__ a_s, float* __restrict__ a_d,
                           int n_nodes) {
    int idx = blockIdx.x * blockDim.x + threadIdx.x;   // n*4 + h
    if (idx >= n_nodes * 4) return;
    int n = idx >> 2, h = idx & 3;
    const float* v  = xp + (size_t)n * 128 + h * 32;
    const float* as = att_src + h * 32;
    const float* ad = att_dst + h * 32;
    float s = 0.0f, d = 0.0f;
    #pragma unroll 8
    for (int c = 0; c < 32; ++c) { float t = v[c]; s += t * as[c]; d += t * ad[c]; }
    a_s[idx] = s; a_d[idx] = d;
}

// ---------------------------------------------------------------------
// edge helpers: edges [0,E) from edge_index, [E, E+N) are self-loops
// ---------------------------------------------------------------------
__device__ __forceinline__ void edge_sd(const int* ei, int e, int E, int& src, int& dst) {
    if (e < E) { src = ei[e]; dst = ei[E + e]; }
    else       { src = e - E; dst = e - E; }
}

// 3) segment max of leaky-relu scores
__global__ void gat_edge_max(const int* __restrict__ ei, int E, int n_nodes,
                             const float* __restrict__ a_s, const float* __restrict__ a_d,
                             int* __restrict__ emax) {
    int e = blockIdx.x * blockDim.x + threadIdx.x;
    if (e >= E + n_nodes) return;
    int src, dst; edge_sd(ei, e, E, src, dst);
    #pragma unroll
    for (int h = 0; h < 4; ++h) {
        float v = leaky(a_s[src * 4 + h] + a_d[dst * 4 + h]);
        atomicMax(&emax[dst * 4 + h], enc_f(v));
    }
}

// 4) segment sum of exp(score - max)
__global__ void gat_edge_denom(const int* __restrict__ ei, int E, int n_nodes,
                               const float* __restrict__ a_s, const float* __restrict__ a_d,
                               const int* __restrict__ emax, float* __restrict__ denom) {
    int e = blockIdx.x * blockDim.x + threadIdx.x;
    if (e >= E + n_nodes) return;
    int src, dst; edge_sd(ei, e, E, src, dst);
    #pragma unroll
    for (int h = 0; h < 4; ++h) {
        float v = leaky(a_s[src * 4 + h] + a_d[dst * 4 + h]);
        float m = dec_f(emax[dst * 4 + h]);
        atomicAdd(&denom[dst * 4 + h], __expf(v - m));
    }
}

// 5) weighted scatter-aggregate: one wave32 per edge, 4 channels per lane
__global__ void gat_edge_agg(const int* __restrict__ ei, int E, int n_nodes,
                             const float* __restrict__ a_s, const float* __restrict__ a_d,
                             const int* __restrict__ emax, const float* __restrict__ denom,
                             const float* __restrict__ xp, float* __restrict__ out) {
    long long gid = (long long)blockIdx.x * blockDim.x + threadIdx.x;
    int e    = (int)(gid >> 5);
    int lane = (int)(gid & 31);
    if (e >= E + n_nodes) return;
    int src, dst; edge_sd(ei, e, E, src, dst);

    int c0 = lane * 4;          // this lane's 4 contiguous channels
    int h  = lane >> 3;         // head = c0 / 32
    float v = leaky(a_s[src * 4 + h] + a_d[dst * 4 + h]);
    float m = dec_f(emax[dst * 4 + h]);
    float alpha = __expf(v - m) / (denom[dst * 4 + h] + 1e-16f);

    const float4 val = *(const float4*)(xp + (size_t)src * 128 + c0);
    float* o = out + (size_t)dst * 128 + c0;
    atomicAdd(o + 0, val.x * alpha);
    atomicAdd(o + 1, val.y * alpha);
    atomicAdd(o + 2, val.z * alpha);
    atomicAdd(o + 3, val.w * alpha);
}

// ---------------------------------------------------------------------
extern "C" void kernel_launch(void* const* d_in, const int* in_sizes, int n_in,
                              void* d_out, int out_size, void* d_ws, size_t ws_size,
                              hipStream_t stream) {
    const float* x       = (const float*)d_in[0];   // [N,128]
    const int*   ei      = (const int*)  d_in[1];   // [2,E]
    const float* lin_w   = (const float*)d_in[2];   // [128,128]
    const float* att_src = (const float*)d_in[3];   // [4,32]
    const float* att_dst = (const float*)d_in[4];   // [4,32]
    const float* bias    = (const float*)d_in[5];   // [128]

    const int n    = in_sizes[0] / 128;
    const int E    = in_sizes[1] / 2;
    const int etot = E + n;

    char* ws = (char*)d_ws;
    float* xp    = (float*)ws; ws += (size_t)n * 128 * sizeof(float);
    float* a_s   = (float*)ws; ws += (size_t)n * 4 * sizeof(float);
    float* a_d   = (float*)ws; ws += (size_t)n * 4 * sizeof(float);
    int*   emax  = (int*)  ws; ws += (size_t)n * 4 * sizeof(int);
    float* denom = (float*)ws;
    float* out   = (float*)d_out;

    const int T = 256;
    gat_init<<<(n * 128 + T - 1) / T, T, 0, stream>>>(out, bias, n * 128, emax, denom, n * 4);
    gat_gemm_wmma<<<(n + 15) / 16, T, 0, stream>>>(x, lin_w, xp, n);
    gat_logits<<<(n * 4 + T - 1) / T, T, 0, stream>>>(xp, att_src, att_dst, a_s, a_d, n);
    gat_edge_max<<<(etot + T - 1) / T, T, 0, stream>>>(ei, E, n, a_s, a_d, emax);
    gat_edge_denom<<<(etot + T - 1) / T, T, 0, stream>>>(ei, E, n, a_s, a_d, emax, denom);
    long long aggthreads = (long long)etot * 32;
    gat_edge_agg<<<(unsigned)((aggthreads + T - 1) / T), T, 0, stream>>>(
        ei, E, n, a_s, a_d, emax, denom, xp, out);
}